// GCNEncoder_74466142978136
// MI455X (gfx1250) — compile-verified
//
#include <hip/hip_runtime.h>

#define HID 64

typedef float v2f __attribute__((ext_vector_type(2)));
typedef float v8f __attribute__((ext_vector_type(8)));

// ---------------------------------------------------------------- zero fill
__global__ void k_zero(float* __restrict__ p, int n) {
  int t = blockIdx.x * blockDim.x + threadIdx.x;
  if (t < n) p[t] = 0.0f;
}

// ---------------------------------------------------------------- degree: deg[dst] += 1 per edge
__global__ void k_degree(const int* __restrict__ ei, int E, float* __restrict__ deg) {
  int e = blockIdx.x * blockDim.x + threadIdx.x;
  if (e < E) atomicAdd(&deg[ei[E + e]], 1.0f);
}

// ---------------------------------------------------------------- dinv = rsqrt(deg + 1 self loop)
__global__ void k_dinv(float* __restrict__ deg, int N) {
  int i = blockIdx.x * blockDim.x + threadIdx.x;
  if (i < N) deg[i] = rsqrtf(deg[i] + 1.0f);
}

// ---------------------------------------------------------------- GEMM1: [N,3] @ [3,64] (memory bound, scalar)
__global__ void k_gemm1(const float* __restrict__ x, const float* __restrict__ W1,
                        float* __restrict__ out, int N) {
  int t = blockIdx.x * blockDim.x + threadIdx.x;
  if (t >= N * HID) return;
  int n = t >> 6, j = t & 63;
  float x0 = x[n * 3 + 0], x1 = x[n * 3 + 1], x2 = x[n * 3 + 2];
  out[t] = x0 * W1[j] + x1 * W1[HID + j] + x2 * W1[2 * HID + j];
}

// ---------------------------------------------------------------- GEMM2: [N,64] @ [64,64] via V_WMMA_F32_16X16X4_F32
// One wave (32 lanes) per 16x16 output tile; K loop 64 in steps of 4.
__global__ __launch_bounds__(32) void k_gemm2(const float* __restrict__ H,
                                              const float* __restrict__ W,
                                              float* __restrict__ O, int N) {
  const int tm   = blockIdx.x;          // M tile index
  const int tn   = blockIdx.y;          // N tile (0..3)
  const int lane = threadIdx.x;         // 0..31, full wave
  const int m    = lane & 15;
  const int half = lane >> 4;           // 0 or 1 -> K sub-pair
  const int ncol = tn * 16 + (lane & 15);

  int arow = tm * 16 + m;
  if (arow >= N) arow = N - 1;          // clamp keeps EXEC full for WMMA
  const float* __restrict__ Arow = H + (long)arow * HID;

  v8f acc = {};
  #pragma unroll
  for (int k0 = 0; k0 < HID; k0 += 4) {
    const int ka = k0 + half * 2;
    v2f a; a.x = Arow[ka];             a.y = Arow[ka + 1];
    v2f b; b.x = W[ka * HID + ncol];   b.y = W[(ka + 1) * HID + ncol];
    acc = __builtin_amdgcn_wmma_f32_16x16x4_f32(
        /*neg_a=*/false, a, /*neg_b=*/false, b,
        /*c_mod=*/(short)0, acc, /*reuse_a=*/false, /*reuse_b=*/false);
  }
  // D layout: VGPR r -> M = r + 8*half, N = lane&15
  #pragma unroll
  for (int r = 0; r < 8; ++r) {
    int mrow = tm * 16 + r + half * 8;
    if (mrow < N) O[(long)mrow * HID + ncol] = acc[r];
  }
}

// ---------------------------------------------------------------- edge scatter: agg[dst] += hw[src]*dinv[s]*dinv[d]
// 16 threads/edge, float4 per thread -> coalesced 256B row reads, L2-resident atomics.
__global__ void k_scatter(const int* __restrict__ ei, int E,
                          const float* __restrict__ hw,
                          const float* __restrict__ dinv,
                          float* __restrict__ agg) {
  int t = blockIdx.x * blockDim.x + threadIdx.x;
  if (t >= E * 16) return;
  int e = t >> 4;
  int j = (t & 15) * 4;
  int s = ei[e];
  int d = ei[E + e];
  float norm = dinv[s] * dinv[d];
  const float4 v = *(const float4*)(hw + (long)s * HID + j);
  float* o = agg + (long)d * HID + j;
  atomicAdd(o + 0, v.x * norm);
  atomicAdd(o + 1, v.y * norm);
  atomicAdd(o + 2, v.z * norm);
  atomicAdd(o + 3, v.w * norm);
}

// ---------------------------------------------------------------- epilogue: h = relu(agg + hw*dinv^2 + b)
// (self-loop message folded in here, not in the atomic path)
__global__ void k_combine(float* __restrict__ agg, const float* __restrict__ hw,
                          const float* __restrict__ dinv, const float* __restrict__ bias,
                          int N) {
  int t = blockIdx.x * blockDim.x + threadIdx.x;
  if (t >= N * 16) return;
  int n = t >> 4;
  int j = (t & 15) * 4;
  float di = dinv[n];
  float sl = di * di;
  float4 a = *(float4*)(agg + (long)n * HID + j);
  const float4 h = *(const float4*)(hw + (long)n * HID + j);
  a.x = fmaxf(fmaf(h.x, sl, a.x) + bias[j + 0], 0.0f);
  a.y = fmaxf(fmaf(h.y, sl, a.y) + bias[j + 1], 0.0f);
  a.z = fmaxf(fmaf(h.z, sl, a.z) + bias[j + 2], 0.0f);
  a.w = fmaxf(fmaf(h.w, sl, a.w) + bias[j + 3], 0.0f);
  *(float4*)(agg + (long)n * HID + j) = a;
}

// ---------------------------------------------------------------- mean pool: sums + counts
__global__ void k_pool(const float* __restrict__ h, const int* __restrict__ batch,
                       float* __restrict__ out, float* __restrict__ cnt, int N) {
  int t = blockIdx.x * blockDim.x + threadIdx.x;
  if (t >= N * 16) return;
  int n = t >> 4;
  int j = (t & 15) * 4;
  int g = batch[n];
  const float4 v = *(const float4*)(h + (long)n * HID + j);
  float* o = out + (long)g * HID + j;
  atomicAdd(o + 0, v.x);
  atomicAdd(o + 1, v.y);
  atomicAdd(o + 2, v.z);
  atomicAdd(o + 3, v.w);
  if ((t & 15) == 0) atomicAdd(&cnt[g], 1.0f);
}

__global__ void k_div(float* __restrict__ out, const float* __restrict__ cnt, int total) {
  int t = blockIdx.x * blockDim.x + threadIdx.x;
  if (t < total) out[t] /= fmaxf(cnt[t >> 6], 1.0f);
}

// ================================================================ launch
extern "C" void kernel_launch(void* const* d_in, const int* in_sizes, int n_in,
                              void* d_out, int out_size, void* d_ws, size_t ws_size,
                              hipStream_t stream) {
  const float* x   = (const float*)d_in[0];
  const float* W1  = (const float*)d_in[1];
  const float* b1  = (const float*)d_in[2];
  const float* W2  = (const float*)d_in[3];
  const float* b2  = (const float*)d_in[4];
  const int*   ei  = (const int*)d_in[5];   // [2,E] flat: row0=src, row1=dst
  const int*   bat = (const int*)d_in[6];
  float* out = (float*)d_out;

  const int N = in_sizes[0] / 3;
  const int E = in_sizes[5] / 2;

  char* ws = (char*)d_ws;
  float* dinv = (float*)(ws);                               // N floats
  float* bufA = (float*)(ws + (1 << 19));                   // N*64 floats (hw)
  float* bufB = (float*)(ws + (1 << 19) + (size_t)N * HID * 4);  // N*64 floats (agg/h)
  float* cnt  = (float*)(ws + (1 << 19) + 2 * (size_t)N * HID * 4);

  const int B = 256;
  const int NH   = N * HID;          // 6.4M
  const int N16  = N * 16;
  const int E16  = E * 16;
  const int tilesM = (N + 15) / 16;

  // normalization
  k_zero  <<<(N + B - 1) / B, B, 0, stream>>>(dinv, N);
  k_degree<<<(E + B - 1) / B, B, 0, stream>>>(ei, E, dinv);
  k_dinv  <<<(N + B - 1) / B, B, 0, stream>>>(dinv, N);

  // ---- layer 1
  k_gemm1  <<<(NH + B - 1) / B, B, 0, stream>>>(x, W1, bufA, N);
  k_zero   <<<(NH + B - 1) / B, B, 0, stream>>>(bufB, NH);
  k_scatter<<<(E16 + B - 1) / B, B, 0, stream>>>(ei, E, bufA, dinv, bufB);
  k_combine<<<(N16 + B - 1) / B, B, 0, stream>>>(bufB, bufA, dinv, b1, N);

  // ---- layer 2 (WMMA GEMM)
  {
    dim3 grid(tilesM, HID / 16);
    k_gemm2<<<grid, 32, 0, stream>>>(bufB, W2, bufA, N);
  }
  k_zero   <<<(NH + B - 1) / B, B, 0, stream>>>(bufB, NH);
  k_scatter<<<(E16 + B - 1) / B, B, 0, stream>>>(ei, E, bufA, dinv, bufB);
  k_combine<<<(N16 + B - 1) / B, B, 0, stream>>>(bufB, bufA, dinv, b2, N);

  // ---- mean pool
  k_zero<<<(out_size + B - 1) / B, B, 0, stream>>>(out, out_size);
  k_zero<<<(512 + B - 1) / B, B, 0, stream>>>(cnt, 512);
  k_pool<<<(N16 + B - 1) / B, B, 0, stream>>>(bufB, bat, out, cnt, N);
  k_div <<<(out_size + B - 1) / B, B, 0, stream>>>(out, cnt, out_size);
}